// Local2Global_120259084729
// MI455X (gfx1250) — compile-verified
//
#include <hip/hip_runtime.h>
#include <hip/hip_bf16.h>

typedef float v2f __attribute__((ext_vector_type(2)));
typedef float v8f __attribute__((ext_vector_type(8)));

#define BS   64
#define CCH  256
#define HWN  3136
#define TT   6
#define DD   192
#define NH   4
#define DH   64
#define NTILES 196          // HWN / 16
#define KSTRIDE 20          // padded LDS row stride (floats): conflict-free both ways
#define NWAVES 16
#define BLKT   512

// ---- gfx1250 async global->LDS path (probe-guarded; fallback = reg prefetch)
#if defined(__has_builtin)
#if __has_builtin(__builtin_amdgcn_global_load_async_to_lds_b128)
#define HAVE_ASYNC_LDS 1
#endif
#endif

#ifdef HAVE_ASYNC_LDS
// Toolchain signature (from diagnostics): (global int4*, local int4*, imm, imm)
typedef int v4i_ __attribute__((vector_size(16)));
typedef __attribute__((address_space(1))) v4i_ GV4;
typedef __attribute__((address_space(3))) v4i_ LV4;
#if __has_builtin(__builtin_amdgcn_s_wait_asynccnt)
#define WAIT_ASYNC(n) __builtin_amdgcn_s_wait_asynccnt((unsigned short)(n))
#else
#define WAIT_ASYNC(n) asm volatile("s_wait_asynccnt %0" ::"n"(n) : "memory")
#endif
#endif

__device__ __forceinline__ v8f wmma_f32(v2f a, v2f b, v8f c) {
  // D(16x16,f32) = A(16x4,f32) * B(4x16,f32) + C
  return __builtin_amdgcn_wmma_f32_16x16x4_f32(false, a, false, b, (short)0, c,
                                               false, false);
}

// ---------------- Stage A: q = tok*q_w^T + q_b ; alpha gate -----------------
__global__ __launch_bounds__(256) void k_qalpha(
    const float* __restrict__ tokens, const float* __restrict__ q_w,
    const float* __restrict__ q_b, const float* __restrict__ a_w,
    const float* __restrict__ a_b, float* __restrict__ q_ws,
    float* __restrict__ alp_ws) {
  const int tb = blockIdx.x;            // t*64 + b
  const int t = tb >> 6, b = tb & 63;
  __shared__ float tok_s[DD];
  if (threadIdx.x < DD) tok_s[threadIdx.x] = tokens[(size_t)tb * DD + threadIdx.x];
  __syncthreads();
  const int c = threadIdx.x;            // 0..255
  float qv = q_b[c], av = a_b[c];
  const float* qr = q_w + (size_t)c * DD;
  const float* ar = a_w + (size_t)c * DD;
#pragma unroll 4
  for (int k = 0; k < DD; ++k) {
    const float x = tok_s[k];
    qv = fmaf(qr[k], x, qv);
    av = fmaf(ar[k], x, av);
  }
  const int head = c >> 6, d = c & 63;
  q_ws[(((size_t)b * NH + head) * TT + t) * DH + d] = qv;   // [b][h][t][d]
  const float g = fminf(fmaxf(av + 3.f, 0.f), 6.f) * (2.f / 6.f);  // h_sigmoid*2
  alp_ws[(size_t)tb * CCH + c] = g;                          // [t][b][c]
}

// -------- Stage B: fused logits + online softmax + attn_out + mlp ----------
// One block per (b, head); 16 waves, each wave owns a strided set of 16-wide
// HW tiles with private flash-attention state; async double-buffered LDS
// staging of the k tile; split-K combine in LDS at the end.
__global__ __launch_bounds__(BLKT) void k_fused(
    const float* __restrict__ feat, const float* __restrict__ mlp_w,
    const float* __restrict__ mlp_b, const float* __restrict__ q_ws,
    const float* __restrict__ alp_ws, float* __restrict__ attn_g,
    float* __restrict__ ts_ws) {
  __shared__ float qs[16][DH];                      // q rows t<6, zero padded
  __shared__ float ktile[NWAVES][2 * DH * KSTRIDE]; // per-wave double buffer
  __shared__ float cAP[NWAVES][TT][DH];             // per-wave attn accumulators
  __shared__ float cAM[NWAVES][TT][DH];             // per-wave mlp accumulators
  __shared__ float cm[NWAVES][8], cl[NWAVES][8];    // per-wave softmax (m,l)

  const int blk = blockIdx.x;
  const int b = blk >> 2, head = blk & 3;
  const int wave = threadIdx.x >> 5;
  const int lane = threadIdx.x & 31;
  const int hi = lane >> 4;                         // lane half
  const int lm = lane & 15;

  const float* kbase = feat + ((size_t)b * CCH + head * DH) * HWN; // k[d][n]

  for (int idx = threadIdx.x; idx < 16 * DH; idx += BLKT) {
    const int t = idx >> 6, d = idx & 63;
    float v = 0.f;
    if (t < TT) v = q_ws[(((size_t)b * NH + head) * TT + t) * DH + d];
    qs[t][d] = v;
  }
  __syncthreads();

  // Pre-build q^T B-fragments for all 16 K-chunks: B[k=kc*4+hi*2+{0,1}][n=t=lm]
  float bq0[16], bq1[16];
#pragma unroll
  for (int kc = 0; kc < 16; ++kc) {
    const int d0 = kc * 4 + hi * 2;
    bq0[kc] = qs[lm][d0];
    bq1[kc] = qs[lm][d0 + 1];
  }

  const v8f vzero = {0.f, 0.f, 0.f, 0.f, 0.f, 0.f, 0.f, 0.f};
  v8f accP[4], accM[4];
#pragma unroll
  for (int i = 0; i < 4; ++i) { accP[i] = vzero; accM[i] = vzero; }
  float m_run = -3.0e38f, l_run = 0.f;

  float* ks = ktile[wave];
  float* outg = attn_g + (((size_t)b * NH + head) * TT) * HWN;

  // ---- per-tile compute from an LDS buffer ----
  auto compute_tile = [&](int nbase, const float* kb) {
    // GEMM-1: L^T[n(16) x t(16)] = k^T * q^T over K=dh (16 chunks)
    v8f cf = vzero;
#pragma unroll
    for (int kc = 0; kc < 16; ++kc) {
      const int d0 = kc * 4 + hi * 2;
      v2f a;  a.x = kb[d0 * KSTRIDE + lm];  a.y = kb[(d0 + 1) * KSTRIDE + lm];
      v2f bq; bq.x = bq0[kc];               bq.y = bq1[kc];
      cf = wmma_f32(a, bq, cf);
    }
#pragma unroll
    for (int r = 0; r < 8; ++r) cf[r] *= 0.125f;   // dh^-0.5

    // Lane holds col t=lm, rows n = nbase + hi*8 + r -> 8 consecutive floats
    if (lm < TT) {
      float* po = outg + (size_t)lm * HWN + nbase + hi * 8;
      *(float4*)(po)     = make_float4(cf[0], cf[1], cf[2], cf[3]);
      *(float4*)(po + 4) = make_float4(cf[4], cf[5], cf[6], cf[7]);
    }

    // online softmax update (per-lane state for t = lm)
    float vmax = cf[0];
#pragma unroll
    for (int r = 1; r < 8; ++r) vmax = fmaxf(vmax, cf[r]);
    const float mt = fmaxf(vmax, __shfl_xor(vmax, 16, 32));
    const float m_new = fmaxf(m_run, mt);
    float p[8]; float s = 0.f;
#pragma unroll
    for (int r = 0; r < 8; ++r) { p[r] = __expf(cf[r] - m_new); s += p[r]; }
    s += __shfl_xor(s, 16, 32);
    const float factor = __expf(m_run - m_new);
    l_run = l_run * factor + s;
    m_run = m_new;
    float fr[8];
#pragma unroll
    for (int r = 0; r < 8; ++r) fr[r] = __shfl(factor, r + 8 * hi, 32);
#pragma unroll
    for (int dt = 0; dt < 4; ++dt)
#pragma unroll
      for (int r = 0; r < 8; ++r) accP[dt][r] *= fr[r];

    // GEMM-2/3: acc[t x d] += P * k^T  and  macc[t x d] += mlp_w * k^T
    const float* mwb = mlp_w + nbase;
#pragma unroll
    for (int kc = 0; kc < 4; ++kc) {
      const int na = kc * 4;
      const int ra = na & 7, rb = (na + 2) & 7;    // src reg per dest half
      const int sl = lm + ((na >= 8) ? 16 : 0);    // src lane
      const float x0a = __shfl(p[ra], sl, 32);
      const float x0b = __shfl(p[rb], sl, 32);
      const float x1a = __shfl(p[ra + 1], sl, 32);
      const float x1b = __shfl(p[rb + 1], sl, 32);
      v2f ap; ap.x = hi ? x0b : x0a; ap.y = hi ? x1b : x1a;
      const int noff = na + hi * 2;
      v2f am;
      if (lm < TT) {
        am.x = mwb[(size_t)lm * HWN + noff];
        am.y = mwb[(size_t)lm * HWN + noff + 1];
      } else { am.x = 0.f; am.y = 0.f; }
#pragma unroll
      for (int dt = 0; dt < 4; ++dt) {
        const int dloc = dt * 16 + lm;
        v2f bf;
        bf.x = kb[dloc * KSTRIDE + noff];
        bf.y = kb[dloc * KSTRIDE + noff + 1];
        accP[dt] = wmma_f32(ap, bf, accP[dt]);
        accM[dt] = wmma_f32(am, bf, accM[dt]);
      }
    }
  };

#ifdef HAVE_ASYNC_LDS
  // ---- async double-buffered staging: DMA next tile while computing ----
  auto issue_tile = [&](int tile, int buf) {
    const int nbase = tile * 16;
    float* dst = ks + buf * (DH * KSTRIDE);
#pragma unroll
    for (int i = 0; i < 8; ++i) {
      const int idx = lane + 32 * i;
      const int r = idx >> 2, quad = idx & 3;
      const float* g = kbase + (size_t)r * HWN + nbase + quad * 4;
      float* l = dst + r * KSTRIDE + quad * 4;
      __builtin_amdgcn_global_load_async_to_lds_b128((GV4*)g, (LV4*)l, 0, 0);
    }
  };
  int buf = 0;
  issue_tile(wave, buf);
  for (int tile = wave; tile < NTILES; tile += NWAVES) {
    const int nxt = tile + NWAVES;
    if (nxt < NTILES) {
      issue_tile(nxt, buf ^ 1);
      WAIT_ASYNC(8);                 // current buffer's 8 transfers complete
    } else {
      WAIT_ASYNC(0);
    }
    compute_tile(tile * 16, ks + buf * (DH * KSTRIDE));
    buf ^= 1;
  }
#else
  // ---- fallback: software-pipelined register prefetch, single buffer ----
  float4 pf[8];
  auto load_regs = [&](int tile) {
    const int nbase = tile * 16;
#pragma unroll
    for (int i = 0; i < 8; ++i) {
      const int idx = lane + 32 * i;
      const int r = idx >> 2, quad = idx & 3;
      pf[i] = *(const float4*)(kbase + (size_t)r * HWN + nbase + quad * 4);
    }
  };
  auto store_regs = [&]() {
#pragma unroll
    for (int i = 0; i < 8; ++i) {
      const int idx = lane + 32 * i;
      const int r = idx >> 2, quad = idx & 3;
      *(float4*)(ks + r * KSTRIDE + quad * 4) = pf[i];
    }
  };
  load_regs(wave);
  for (int tile = wave; tile < NTILES; tile += NWAVES) {
    store_regs();                    // DS is in-order: safe vs prior tile reads
    const int nxt = tile + NWAVES;
    if (nxt < NTILES) load_regs(nxt);  // overlap next HBM fetch with compute
    compute_tile(tile * 16, ks);
  }
#endif

  // ---- split-K combine across the waves ----
  if (hi == 0 && lm < TT) { cm[wave][lm] = m_run; cl[wave][lm] = l_run; }
  if (hi == 0) {                  // lanes 0..15 hold rows t = r (r < 6 valid)
#pragma unroll
    for (int dt = 0; dt < 4; ++dt)
#pragma unroll
      for (int r = 0; r < TT; ++r) {
        cAP[wave][r][dt * 16 + lm] = accP[dt][r];
        cAM[wave][r][dt * 16 + lm] = accM[dt][r];
      }
  }
  __syncthreads();

  for (int idx = threadIdx.x; idx < TT * DH; idx += BLKT) {
    const int t = idx >> 6, d = idx & 63;
    float M = cm[0][t];
#pragma unroll
    for (int w = 1; w < NWAVES; ++w) M = fmaxf(M, cm[w][t]);
    float L = 0.f, P = 0.f, Q = 0.f;
#pragma unroll
    for (int w = 0; w < NWAVES; ++w) {
      const float e = __expf(cm[w][t] - M);
      L += cl[w][t] * e;
      P += cAP[w][t][d] * e;
      Q += cAM[w][t][d];
    }
    const int cch = head * DH + d;
    const float tsv = (Q + mlp_b[t] + P / L) * alp_ws[((size_t)t * BS + b) * CCH + cch];
    ts_ws[((size_t)t * BS + b) * CCH + cch] = tsv;
  }
}

// ---------------- Stage C: proj + residual + LayerNorm ---------------------
__global__ __launch_bounds__(192) void k_projln(
    const float* __restrict__ ts_ws, const float* __restrict__ tokens,
    const float* __restrict__ proj_w, const float* __restrict__ proj_b,
    const float* __restrict__ ln_g, const float* __restrict__ ln_b,
    float* __restrict__ tok_out) {
  const int row = blockIdx.x;           // t*64 + b
  __shared__ float ts_s[CCH];
  __shared__ float rs[6], rq[6];
  for (int idx = threadIdx.x; idx < CCH; idx += 192)
    ts_s[idx] = ts_ws[(size_t)row * CCH + idx];
  __syncthreads();
  const int j = threadIdx.x;
  float y = proj_b[j];
  const float* pw = proj_w + (size_t)j * CCH;
#pragma unroll 4
  for (int k = 0; k < CCH; ++k) y = fmaf(pw[k], ts_s[k], y);
  y += tokens[(size_t)row * DD + j];
  float s = y, q = y * y;
#pragma unroll
  for (int off = 16; off > 0; off >>= 1) {
    s += __shfl_down(s, off, 32);
    q += __shfl_down(q, off, 32);
  }
  if ((threadIdx.x & 31) == 0) { rs[threadIdx.x >> 5] = s; rq[threadIdx.x >> 5] = q; }
  __syncthreads();
  float ss = 0.f, qq = 0.f;
#pragma unroll
  for (int w = 0; w < 6; ++w) { ss += rs[w]; qq += rq[w]; }
  const float mu = ss / DD;
  const float var = qq / DD - mu * mu;
  const float rr = rsqrtf(var + 1e-5f);
  tok_out[(size_t)row * DD + j] = (y - mu) * rr * ln_g[j] + ln_b[j];
}

extern "C" void kernel_launch(void* const* d_in, const int* in_sizes, int n_in,
                              void* d_out, int out_size, void* d_ws, size_t ws_size,
                              hipStream_t stream) {
  (void)in_sizes; (void)n_in; (void)out_size; (void)ws_size;
  const float* features = (const float*)d_in[0];
  const float* tokens   = (const float*)d_in[1];
  const float* mlp_w    = (const float*)d_in[2];
  const float* mlp_b    = (const float*)d_in[3];
  const float* q_w      = (const float*)d_in[4];
  const float* q_b      = (const float*)d_in[5];
  const float* alpha_w  = (const float*)d_in[6];
  const float* alpha_b  = (const float*)d_in[7];
  const float* proj_w   = (const float*)d_in[8];
  const float* proj_b   = (const float*)d_in[9];
  const float* ln_g     = (const float*)d_in[10];
  const float* ln_b     = (const float*)d_in[11];

  float* tok_out  = (float*)d_out;                // [6,64,192]
  float* attn_out = tok_out + TT * BS * DD;       // [64,4,6,3136]

  float* ws     = (float*)d_ws;
  float* q_ws   = ws;                             // 98304 f
  float* alp_ws = ws + 98304;                     // 98304 f
  float* ts_ws  = ws + 2 * 98304;                 // 98304 f

  k_qalpha<<<TT * BS, 256, 0, stream>>>(tokens, q_w, q_b, alpha_w, alpha_b,
                                        q_ws, alp_ws);
  k_fused<<<BS * NH, BLKT, 0, stream>>>(features, mlp_w, mlp_b, q_ws, alp_ws,
                                        attn_out, ts_ws);
  k_projln<<<TT * BS, 192, 0, stream>>>(ts_ws, tokens, proj_w, proj_b,
                                        ln_g, ln_b, tok_out);
}